// MultiheadAttention_58729382805511
// MI455X (gfx1250) — compile-verified
//
#include <hip/hip_runtime.h>
#include <math.h>
#include <stdint.h>

// Problem constants (from reference): B=16, S=1 decode step, GQA 32q/8kv heads.
#define B_    16
#define D_    4096
#define H_    32
#define KVH_  8
#define HD_   128
#define R_    4          // H/KVH
#define MAXS_ 4096
#define NQ_   (H_*HD_)   // 4096
#define NK_   (KVH_*HD_) // 1024
#define NV_   (KVH_*HD_) // 1024
#define NTOT_ (NQ_+NK_+NV_) // 6144
#define NCHUNK_ 8
#define TCHUNK_ 512
#define KROWS_  64                    // K rows per LDS subtile
#define NSUB_   (TCHUNK_/KROWS_)      // 8 subtiles per chunk
#define KPAD_   132                   // padded row stride (floats) -> bank spread
#define SCALE_ 0.08838834764831845f   // 1/sqrt(128)

typedef float v2f __attribute__((ext_vector_type(2)));
typedef float v8f __attribute__((ext_vector_type(8)));

// ---------------------------------------------------------------------------
// CDNA5 async copy: global -> LDS, 16B per lane, tracked on ASYNCcnt.
// lds_off = byte offset within the workgroup LDS allocation (low 32 bits of a
// generic LDS pointer), gaddr = 64-bit global address.
// ---------------------------------------------------------------------------
__device__ __forceinline__ void async_load_b128(unsigned lds_off, const void* gaddr) {
  asm volatile("global_load_async_to_lds_b128 %0, %1, off"
               :: "v"(lds_off), "v"(gaddr)
               : "memory");
}

// ---------------------------------------------------------------------------
// One wave computes a 16(M=batch rows) x 16(N=cols) fp32 tile of A(16xK) @ W(KxN)
// with V_WMMA_F32_16X16X4_F32, looping K in steps of 4.
// A-layout (16x4 f32): lanes0-15 hold M=lane, VGPR0/1 = K={0,1}; lanes16-31 K={2,3}.
// B-layout (4x16 f32): VGPR0: lanes0-15 row K=0 (N=lane), lanes16-31 row K=2;
//                      VGPR1: rows K=1 / K=3.
// C-layout: VGPR i: lanes0-15 -> (M=i, N=lane); lanes16-31 -> (M=8+i, N=lane-16).
// ---------------------------------------------------------------------------
__device__ __forceinline__ v8f wmma_rowtile(const float* __restrict__ A, int lda,
                                            const float* __restrict__ W, int ldw,
                                            int K, int col) {
  const int lane = threadIdx.x & 31;
  const int half = lane >> 4;
  v8f c = {};
  const float* arow = A + (size_t)(lane & 15) * lda;   // M = lane%16
  for (int k0 = 0; k0 < K; k0 += 4) {
    const int ka = k0 + 2 * half;
    const v2f a = *(const v2f*)(arow + ka);
    v2f b;
    b.x = W[(size_t)ka       * ldw + col];
    b.y = W[(size_t)(ka + 1) * ldw + col];
    if ((k0 & 31) == 0)  // -> global_prefetch_b8, keep the weight stream hot
      __builtin_prefetch(W + (size_t)(k0 + 32) * ldw + col, 0, 1);
    // 8 args: (neg_a, A, neg_b, B, c_mod, C, reuse_a, reuse_b)
    c = __builtin_amdgcn_wmma_f32_16x16x4_f32(false, a, false, b, (short)0, c,
                                              false, false);
  }
  return c;
}

// ---------------------------------------------------------------------------
// Kernel 1: fused QKV projection + RoPE, scattered to workspace (inputs are
// never mutated; the fresh K/V row is merged at t==start_pos during attention).
// Grid: NTOT_/16 = 384 blocks x 32 threads (1 wave).
// ---------------------------------------------------------------------------
__global__ void __launch_bounds__(32)
qkv_kernel(const float* __restrict__ x,
           const float* __restrict__ wq, const float* __restrict__ wk,
           const float* __restrict__ wv,
           const float* __restrict__ fcos, const float* __restrict__ fsin,
           float* __restrict__ ws_q, float* __restrict__ ws_k,
           float* __restrict__ ws_v) {
  const int n0   = blockIdx.x * 16;
  const int lane = threadIdx.x;
  const int half = lane >> 4, ln = lane & 15;

  const float* W; int ldw; int cb; float* dst; int dstride; bool rope;
  if (n0 < NQ_)            { W = wq; ldw = NQ_; cb = n0;             dst = ws_q; dstride = NQ_; rope = true;  }
  else if (n0 < NQ_ + NK_) { W = wk; ldw = NK_; cb = n0 - NQ_;       dst = ws_k; dstride = NK_; rope = true;  }
  else                     { W = wv; ldw = NV_; cb = n0 - NQ_ - NK_; dst = ws_v; dstride = NV_; rope = false; }
  const int col = cb + ln;

  v8f c = wmma_rowtile(x, D_, W, ldw, D_, col);

  float cs = 1.f, sn = 0.f;
  if (rope) {
    const int fi = (col & (HD_ - 1)) >> 1;  // freq index within head (S=1 row)
    cs = fcos[fi]; sn = fsin[fi];
  }
  const bool odd = (ln & 1) != 0;
#pragma unroll
  for (int i = 0; i < 8; ++i) {
    float v = c[i];
    if (rope) {
      const float p = __shfl_xor(v, 1, 32);  // partner column in the RoPE pair
      // even col: a*cos - b*sin ; odd col: a*sin + b*cos
      v = odd ? __fmaf_rn(p, sn, v * cs) : __fmaf_rn(v, cs, -(p * sn));
    }
    const int m = i + half * 8;              // batch row
    dst[(size_t)m * dstride + col] = v;
  }
}

// ---------------------------------------------------------------------------
// Kernel 2: flash-decoding attention partials.
// Grid: B*KVH*NCHUNK = 1024 blocks x 256 threads; block = (b, g, chunk of 512 t).
// Score pass streams K through LDS via async b128 copies (double-buffered
// 64-row tiles, ASYNCcnt pipelined); P@V pass reads V coalesced from global.
// ---------------------------------------------------------------------------
__global__ void __launch_bounds__(256)
attn_partial_kernel(const float* __restrict__ ws_q, const float* __restrict__ ws_k,
                    const float* __restrict__ ws_v,
                    const float* __restrict__ cache_k,
                    const float* __restrict__ cache_v,
                    const int* __restrict__ start_pos,
                    float* __restrict__ part_acc, float* __restrict__ part_ml) {
  __shared__ float q_s[R_ * HD_];              // 2 KB
  __shared__ float p_s[R_ * TCHUNK_];          // 8 KB
  __shared__ float red[8];
  __shared__ float kt[2][KROWS_ * KPAD_];      // 2 x 33 KB K tiles (padded rows)

  const int tid = threadIdx.x;
  const int idx = blockIdx.x;
  const int ch  = idx & (NCHUNK_ - 1);
  const int bg  = idx >> 3;
  const int g   = bg & (KVH_ - 1);
  const int b   = bg >> 3;
  const int sp  = *start_pos;
  const int T   = sp + 1;
  const int t0  = ch * TCHUNK_;

  const unsigned kt_base = (unsigned)(uintptr_t)&kt[0][0];

  // Stage Q (heads g*R .. g*R+3 are contiguous) into LDS.
  for (int i = tid; i < R_ * HD_; i += 256)
    q_s[i] = ws_q[(size_t)b * NQ_ + (size_t)(g * R_) * HD_ + i];

  // Issue one 64-row K subtile: 2048 x 16B transfers, 8 per thread, coalesced.
  auto issue_tile = [&](int sub, int buf) {
#pragma unroll
    for (int i = 0; i < 8; ++i) {
      const int f   = i * 256 + tid;    // transfer id 0..2047
      const int row = f >> 5;           // 32 x 16B chunks per 512B row
      const int cid = f & 31;
      const int t   = t0 + sub * KROWS_ + row;   // always < MAXS_
      const float* src = (t == sp)
          ? (ws_k + ((size_t)b * KVH_ + g) * HD_ + cid * 4)
          : (cache_k + (((size_t)b * MAXS_ + t) * KVH_ + g) * HD_ + cid * 4);
      const unsigned loff = kt_base +
          (unsigned)(buf * (KROWS_ * KPAD_ * 4) + row * (KPAD_ * 4) + cid * 16);
      async_load_b128(loff, src);
    }
  };

  // ---- score pass: thread = (head qh, row) ----
  const int qh  = tid >> 6;     // 0..3
  const int row = tid & (KROWS_ - 1);
  float sloc[NSUB_];

  issue_tile(0, 0);
#pragma unroll
  for (int sub = 0; sub < NSUB_; ++sub) {
    const int buf = sub & 1;
    if (sub + 1 < NSUB_) {
      issue_tile(sub + 1, buf ^ 1);
      __builtin_amdgcn_s_wait_asynccnt(8);   // current tile retired, next in flight
    } else {
      __builtin_amdgcn_s_wait_asynccnt(0);   // final tile fully landed
    }
    __syncthreads();                          // tile visible to all waves

    const float* kr = &kt[buf][row * KPAD_];
    float acc = 0.f;
#pragma unroll 8
    for (int d4 = 0; d4 < HD_; d4 += 4) {
      const float4 kv = *(const float4*)(kr + d4);
      acc += q_s[qh*HD_+d4  ]*kv.x + q_s[qh*HD_+d4+1]*kv.y
           + q_s[qh*HD_+d4+2]*kv.z + q_s[qh*HD_+d4+3]*kv.w;
    }
    const int t = t0 + sub * KROWS_ + row;
    sloc[sub] = (t < T) ? acc * SCALE_ : -INFINITY;

    __syncthreads();                          // buffer free before reuse
  }

  // ---- chunk-local softmax per head (64 threads = 2 waves per head) ----
  const int wid = tid >> 5, lane = tid & 31;
  float vmax = sloc[0];
#pragma unroll
  for (int i = 1; i < NSUB_; ++i) vmax = fmaxf(vmax, sloc[i]);
  for (int o = 16; o > 0; o >>= 1) vmax = fmaxf(vmax, __shfl_xor(vmax, o, 32));
  if (lane == 0) red[wid] = vmax;
  __syncthreads();
  float m[R_];
#pragma unroll
  for (int r = 0; r < R_; ++r) m[r] = fmaxf(red[2*r], red[2*r + 1]);
  __syncthreads();

  float lsum = 0.f;
#pragma unroll
  for (int sub = 0; sub < NSUB_; ++sub) {
    const float p = (sloc[sub] > -1e37f && m[qh] > -1e37f)
                        ? __expf(sloc[sub] - m[qh]) : 0.f;
    p_s[qh * TCHUNK_ + sub * KROWS_ + row] = p;
    lsum += p;
  }
  for (int o = 16; o > 0; o >>= 1) lsum += __shfl_xor(lsum, o, 32);
  if (lane == 0) red[wid] = lsum;
  __syncthreads();
  float l[R_];
#pragma unroll
  for (int r = 0; r < R_; ++r) l[r] = red[2*r] + red[2*r + 1];

  // ---- P @ V: d-parallel; V loads coalesced 512B per 128 lanes ----
  const int r0 = tid >> 7;          // each thread owns heads r0 and r0+2
  const int d  = tid & (HD_ - 1);
  float a0 = 0.f, a1 = 0.f;
  int tn = T - t0; if (tn > TCHUNK_) tn = TCHUNK_; if (tn < 0) tn = 0;
  for (int t = 0; t < tn; ++t) {
    const int tt = t0 + t;
    const float v = (tt == sp)
        ? ws_v[((size_t)b * KVH_ + g) * HD_ + d]
        : cache_v[(((size_t)b * MAXS_ + tt) * KVH_ + g) * HD_ + d];
    a0 = __fmaf_rn(p_s[r0 * TCHUNK_ + t], v, a0);
    a1 = __fmaf_rn(p_s[(r0 + 2) * TCHUNK_ + t], v, a1);
  }

  const size_t pb = (size_t)idx * (R_ * HD_);
  part_acc[pb + (size_t)r0 * HD_ + d]       = a0;
  part_acc[pb + (size_t)(r0 + 2) * HD_ + d] = a1;
  if (tid < R_) {
    part_ml[(size_t)idx * R_ * 2 + tid * 2 + 0] = m[tid];
    part_ml[(size_t)idx * R_ * 2 + tid * 2 + 1] = l[tid];
  }
}

// ---------------------------------------------------------------------------
// Kernel 3: combine the NCHUNK partials per (b,g) with LSE rescaling.
// Grid: B*KVH = 128 blocks x 128 threads (d).
// ---------------------------------------------------------------------------
__global__ void __launch_bounds__(128)
attn_reduce_kernel(const float* __restrict__ part_acc,
                   const float* __restrict__ part_ml,
                   float* __restrict__ attn_out) {
  const int bg = blockIdx.x;
  const int g  = bg & (KVH_ - 1);
  const int b  = bg >> 3;
  const int d  = threadIdx.x;
#pragma unroll
  for (int r = 0; r < R_; ++r) {
    float M = -INFINITY;
    for (int ch = 0; ch < NCHUNK_; ++ch)
      M = fmaxf(M, part_ml[((size_t)(bg * NCHUNK_ + ch) * R_ + r) * 2 + 0]);
    float L = 0.f, o = 0.f;
    for (int ch = 0; ch < NCHUNK_; ++ch) {
      const int idx  = bg * NCHUNK_ + ch;
      const float mc = part_ml[((size_t)idx * R_ + r) * 2 + 0];
      const float lc = part_ml[((size_t)idx * R_ + r) * 2 + 1];
      const float w  = (mc > -1e37f) ? __expf(mc - M) : 0.f;
      L += w * lc;
      o = __fmaf_rn(w, part_acc[((size_t)idx * R_ + r) * HD_ + d], o);
    }
    attn_out[((size_t)b * H_ + (g * R_ + r)) * HD_ + d] = o / L;
  }
}

// ---------------------------------------------------------------------------
// Kernel 4: output projection attn(16x4096) @ wo(4096x4096) -> d_out, WMMA tiles.
// Grid: D_/16 = 256 blocks x 32 threads.
// ---------------------------------------------------------------------------
__global__ void __launch_bounds__(32)
out_proj_kernel(const float* __restrict__ attn, const float* __restrict__ wo,
                float* __restrict__ out) {
  const int n0   = blockIdx.x * 16;
  const int lane = threadIdx.x;
  const int half = lane >> 4, ln = lane & 15;
  const int col  = n0 + ln;
  v8f c = wmma_rowtile(attn, D_, wo, D_, D_, col);
#pragma unroll
  for (int i = 0; i < 8; ++i)
    out[(size_t)(i + half * 8) * D_ + col] = c[i];
}

// ---------------------------------------------------------------------------
extern "C" void kernel_launch(void* const* d_in, const int* in_sizes, int n_in,
                              void* d_out, int out_size, void* d_ws, size_t ws_size,
                              hipStream_t stream) {
  (void)in_sizes; (void)n_in; (void)out_size; (void)ws_size;
  const float* x       = (const float*)d_in[0];
  const int*   sp      = (const int*)  d_in[1];
  const float* fcos    = (const float*)d_in[2];
  const float* fsin    = (const float*)d_in[3];
  const float* cache_k = (const float*)d_in[4];
  const float* cache_v = (const float*)d_in[5];
  const float* wq      = (const float*)d_in[6];
  const float* wk      = (const float*)d_in[7];
  const float* wv      = (const float*)d_in[8];
  const float* wo      = (const float*)d_in[9];
  float* out = (float*)d_out;

  // Workspace layout (floats): ~2.8 MB total
  float* ws     = (float*)d_ws;
  float* ws_q   = ws;                  // 16*32*128      = 65536
  float* ws_k   = ws_q  + 65536;       // 16*8*128       = 16384
  float* ws_v   = ws_k  + 16384;       // 16*8*128       = 16384
  float* p_acc  = ws_v  + 16384;       // 1024*4*128     = 524288
  float* p_ml   = p_acc + 524288;      // 1024*4*2       = 8192
  float* attn   = p_ml  + 8192;        // 16*32*128      = 65536

  qkv_kernel<<<NTOT_ / 16, 32, 0, stream>>>(x, wq, wk, wv, fcos, fsin,
                                            ws_q, ws_k, ws_v);
  attn_partial_kernel<<<B_ * KVH_ * NCHUNK_, 256, 0, stream>>>(
      ws_q, ws_k, ws_v, cache_k, cache_v, sp, p_acc, p_ml);
  attn_reduce_kernel<<<B_ * KVH_, 128, 0, stream>>>(p_acc, p_ml, attn);
  out_proj_kernel<<<D_ / 16, 32, 0, stream>>>(attn, wo, out);
}